// EdgeEncoder_90761248899724
// MI455X (gfx1250) — compile-verified
//
#include <hip/hip_runtime.h>
#include <hip/hip_bf16.h>

typedef __attribute__((ext_vector_type(2))) float v2f;
typedef __attribute__((ext_vector_type(8))) float v8f;

namespace {
constexpr int G   = 64;          // graphs
constexpr int NN  = 512;         // nodes per graph
constexpr int KN  = 16;          // knn k
constexpr int E   = G * NN * KN; // 524288 edges
constexpr int D1  = 4, D2 = 64, D3 = 128, D4 = 256;
constexpr int H1S = D2 + 1;      // padded LDS row strides (bank-conflict free)
constexpr int H2S = D3 + 1;
constexpr int MLP_WAVES = 4;     // waves per block (wave32)
constexpr int SMEM_FLOATS =
    D1 * D2 + D2 * D3 + D3 * D4 + MLP_WAVES * (16 * H1S + 16 * H2S);
} // namespace

// ---------------------------------------------------------------------------
// Kernel 1: per-graph 2-D KNN (k=16), raw 4-feature edge attributes,
//           and edge_index written (as float) into the tail of d_out.
// ---------------------------------------------------------------------------
__global__ void knn_edge_kernel(const float* __restrict__ loc,
                                float* __restrict__ raw,   // [E,4] workspace
                                float* __restrict__ out)   // full d_out
{
    __shared__ float sx[NN], sy[NN], s4[NN], s5[NN], s8[NN], s9[NN];
    const int g   = blockIdx.x;
    const int tid = threadIdx.x;
    const float* L = loc + (size_t)g * NN * 10;
    for (int i = tid; i < NN; i += blockDim.x) {
        const float* r = L + i * 10;
        s4[i] = r[4]; s5[i] = r[5];
        sx[i] = r[6]; sy[i] = r[7];
        s8[i] = r[8]; s9[i] = r[9];
    }
    __syncthreads();

    for (int r = tid; r < NN; r += blockDim.x) {
        float bd[KN];
        int   bi[KN];
#pragma unroll
        for (int s = 0; s < KN; ++s) { bd[s] = __builtin_inff(); bi[s] = 0; }
        const float qx = sx[r], qy = sy[r];
        for (int j = 0; j < NN; ++j) {
            if (j == r) continue;                 // no self loop
            const float dx = sx[j] - qx, dy = sy[j] - qy;
            const float d  = dx * dx + dy * dy;
            if (d < bd[KN - 1]) {                 // strict <: stable ties (lower idx wins)
                float nd = d; int ni = j;
#pragma unroll
                for (int s = 0; s < KN; ++s) {
                    if (nd < bd[s]) {
                        float td = bd[s]; int ti = bi[s];
                        bd[s] = nd; bi[s] = ni; nd = td; ni = ti;
                    }
                }
            }
        }
        const int node = g * NN + r;                   // target (center) node
        float* rawp = raw + (size_t)node * KN * 4;
        float* srcp = out + (size_t)E * D4 + (size_t)node * KN;
        float* tgtp = srcp + E;
        const float t4 = s4[r], t5 = s5[r];
#pragma unroll
        for (int jj = 0; jj < KN; ++jj) {
            const int j = bi[jj];                      // source (neighbor) node
            rawp[jj * 4 + 0] = (sx[j] - qx) / s8[j];
            rawp[jj * 4 + 1] = (sy[j] - qy) / s9[j];
            rawp[jj * 4 + 2] = __logf(s4[j] / t4);
            rawp[jj * 4 + 3] = __logf(s5[j] / t5);
            srcp[jj] = (float)(g * NN + j);
            tgtp[jj] = (float)node;
        }
    }
}

// ---------------------------------------------------------------------------
// Kernel 2: 4->64->128->256 ReLU MLP over all edges with fp32 WMMA.
// Weights transposed in LDS; per-wave padded LDS tiles re-lay-out the WMMA
// D layout into the next layer's A layout. N-tile loops deliberately NOT
// unrolled (short accumulator live ranges -> no spills); ILP comes from a
// dual-accumulator (two N-tiles per pass) sharing the A-operand loads.
// ---------------------------------------------------------------------------
__global__ void mlp_wmma_kernel(const float* __restrict__ raw,
                                const float* __restrict__ W1, const float* __restrict__ b1,
                                const float* __restrict__ W2, const float* __restrict__ b2,
                                const float* __restrict__ W3, const float* __restrict__ b3,
                                float* __restrict__ out)
{
    extern __shared__ float smem[];
    float* sW1 = smem;                        // [D1][D2]  transposed: sW1[k*D2+n]
    float* sW2 = sW1 + D1 * D2;               // [D2][D3]
    float* sW3 = sW2 + D2 * D3;               // [D3][D4]
    float* act = sW3 + D3 * D4;               // per-wave h1/h2 tiles

    const int tid    = threadIdx.x;
    const int lane   = tid & 31;
    const int wave   = tid >> 5;
    const int nwaves = blockDim.x >> 5;
    const int half   = lane >> 4;             // A/B lane half: K offset 0 or 2
    const int ln     = lane & 15;             // row (A) / column (B,C,D)

    // Cooperative transposed weight load: W[n][k] -> sW[k][n]
    for (int i = tid; i < D1 * D2; i += blockDim.x) { int n = i / D1, k = i % D1; sW1[k * D2 + n] = W1[i]; }
    for (int i = tid; i < D2 * D3; i += blockDim.x) { int n = i / D2, k = i % D2; sW2[k * D3 + n] = W2[i]; }
    for (int i = tid; i < D3 * D4; i += blockDim.x) { int n = i / D3, k = i % D3; sW3[k * D4 + n] = W3[i]; }
    __syncthreads();

    float* h1 = act + wave * (16 * H1S + 16 * H2S);
    float* h2 = h1 + 16 * H1S;

    const int k2h = 2 * half;                 // K base offset within a K4 step
    const int ntile_total = E / 16;
    const int gw     = blockIdx.x * nwaves + wave;
    const int stride = gridDim.x * nwaves;

    for (int t = gw; t < ntile_total; t += stride) {
        const int base = t * 16;

        // ----- layer 1: [16x4] @ [4x64], single K-step, 4 N-tiles -----
        v2f a1 = *(const v2f*)(raw + (size_t)(base + ln) * D1 + k2h);
#pragma unroll 1
        for (int nt = 0; nt < D2 / 16; ++nt) {
            const int col = nt * 16 + ln;
            const float bias = b1[col];
            v8f c;
#pragma unroll
            for (int v = 0; v < 8; ++v) c[v] = bias;
            v2f b;
            b.x = sW1[k2h * D2 + col];
            b.y = sW1[(k2h + 1) * D2 + col];
            c = __builtin_amdgcn_wmma_f32_16x16x4_f32(false, a1, false, b,
                                                      (short)0, c, false, false);
            float* hp = h1 + 8 * half * H1S + col;
#pragma unroll
            for (int v = 0; v < 8; ++v) hp[v * H1S] = fmaxf(c[v], 0.f);
        }

        // ----- layer 2: [16x64] @ [64x128], 8 N-tiles, 2 at a time -----
#pragma unroll 1
        for (int nt = 0; nt < D3 / 16; nt += 2) {
            const int col0 = nt * 16 + ln;
            const int col1 = col0 + 16;
            const float bias0 = b2[col0], bias1 = b2[col1];
            v8f c0, c1;
#pragma unroll
            for (int v = 0; v < 8; ++v) { c0[v] = bias0; c1[v] = bias1; }
#pragma unroll 2
            for (int kk = 0; kk < D2 / 4; ++kk) {
                const int k0 = kk * 4 + k2h;
                v2f a, b0, b1v;
                a.x  = h1[ln * H1S + k0];
                a.y  = h1[ln * H1S + k0 + 1];
                b0.x = sW2[k0 * D3 + col0];
                b0.y = sW2[(k0 + 1) * D3 + col0];
                b1v.x = sW2[k0 * D3 + col1];
                b1v.y = sW2[(k0 + 1) * D3 + col1];
                c0 = __builtin_amdgcn_wmma_f32_16x16x4_f32(false, a, false, b0,
                                                           (short)0, c0, false, false);
                c1 = __builtin_amdgcn_wmma_f32_16x16x4_f32(false, a, false, b1v,
                                                           (short)0, c1, false, false);
            }
            float* hp = h2 + 8 * half * H2S + col0;
#pragma unroll
            for (int v = 0; v < 8; ++v) {
                hp[v * H2S]      = fmaxf(c0[v], 0.f);
                hp[v * H2S + 16] = fmaxf(c1[v], 0.f);
            }
        }

        // ----- layer 3: [16x128] @ [128x256] -> global store, 2 N-tiles/pass --
#pragma unroll 1
        for (int nt = 0; nt < D4 / 16; nt += 2) {
            const int col0 = nt * 16 + ln;
            const int col1 = col0 + 16;
            const float bias0 = b3[col0], bias1 = b3[col1];
            v8f c0, c1;
#pragma unroll
            for (int v = 0; v < 8; ++v) { c0[v] = bias0; c1[v] = bias1; }
#pragma unroll 2
            for (int kk = 0; kk < D3 / 4; ++kk) {
                const int k0 = kk * 4 + k2h;
                v2f a, b0, b1v;
                a.x  = h2[ln * H2S + k0];
                a.y  = h2[ln * H2S + k0 + 1];
                b0.x = sW3[k0 * D4 + col0];
                b0.y = sW3[(k0 + 1) * D4 + col0];
                b1v.x = sW3[k0 * D4 + col1];
                b1v.y = sW3[(k0 + 1) * D4 + col1];
                c0 = __builtin_amdgcn_wmma_f32_16x16x4_f32(false, a, false, b0,
                                                           (short)0, c0, false, false);
                c1 = __builtin_amdgcn_wmma_f32_16x16x4_f32(false, a, false, b1v,
                                                           (short)0, c1, false, false);
            }
            float* op = out + (size_t)(base + 8 * half) * D4 + col0;
#pragma unroll
            for (int v = 0; v < 8; ++v) {
                op[(size_t)v * D4]      = fmaxf(c0[v], 0.f);
                op[(size_t)v * D4 + 16] = fmaxf(c1[v], 0.f);
            }
        }
    }
}

// ---------------------------------------------------------------------------
extern "C" void kernel_launch(void* const* d_in, const int* in_sizes, int n_in,
                              void* d_out, int out_size, void* d_ws, size_t ws_size,
                              hipStream_t stream) {
    const float* loc = (const float*)d_in[0];
    const float* W1  = (const float*)d_in[1];
    const float* b1  = (const float*)d_in[2];
    const float* W2  = (const float*)d_in[3];
    const float* b2  = (const float*)d_in[4];
    const float* W3  = (const float*)d_in[5];
    const float* b3  = (const float*)d_in[6];
    float* out = (float*)d_out;
    float* raw = (float*)d_ws;   // [E,4] raw edge features

    knn_edge_kernel<<<G, 256, 0, stream>>>(loc, raw, out);

    static_assert(SMEM_FLOATS * 4 <= 320 * 1024, "exceeds 320KB WGP LDS");
    (void)hipFuncSetAttribute((const void*)mlp_wmma_kernel,
                              hipFuncAttributeMaxDynamicSharedMemorySize,
                              SMEM_FLOATS * 4);
    mlp_wmma_kernel<<<512, MLP_WAVES * 32, SMEM_FLOATS * 4, stream>>>(
        raw, W1, b1, W2, b2, W3, b3, out);
}